// GraphConvolution_26714696581338
// MI455X (gfx1250) — compile-verified
//
#include <hip/hip_runtime.h>
#include <hip/hip_bf16.h>
#include <stdint.h>

// Problem constants (match reference)
#define Mv   49152
#define Bv   16
#define FIN  16
#define F1   32
#define Kc   5
#define ROWW 256                      // FIN*B columns per working row
static const size_t TERM = (size_t)Mv * ROWW;   // floats per Chebyshev term

typedef __attribute__((ext_vector_type(2))) float v2f;
typedef __attribute__((ext_vector_type(8))) float v8f;

// ---------------------------------------------------------------------------
// Guaranteed-native f32 atomic adds: one VGPR-pair address + immediate offsets,
// no-return form (STOREcnt only), device scope so the RMW is L2-coherent
// across workgroups. Avoids any chance of a CAS-loop expansion of atomicAdd.
// ---------------------------------------------------------------------------
__device__ __forceinline__ void atomic_add_f32x4(float* p, float4 v) {
  asm volatile(
      "global_atomic_add_f32 %0, %1, off scope:SCOPE_DEV\n\t"
      "global_atomic_add_f32 %0, %2, off offset:4 scope:SCOPE_DEV\n\t"
      "global_atomic_add_f32 %0, %3, off offset:8 scope:SCOPE_DEV\n\t"
      "global_atomic_add_f32 %0, %4, off offset:12 scope:SCOPE_DEV"
      :
      : "v"((unsigned long long)(uintptr_t)p),
        "v"(v.x), "v"(v.y), "v"(v.z), "v"(v.w)
      : "memory");
}

// ---------------------------------------------------------------------------
// Pack x [B, M, FIN] -> x0 [M, B*FIN]  (x0[m, b*16+f] = x[b,m,f])
// ---------------------------------------------------------------------------
__global__ __launch_bounds__(256) void cheb_pack(const float* __restrict__ x,
                                                 float* __restrict__ x0) {
  const int m = blockIdx.x;
  const int t = threadIdx.x;          // 0..255
  const int b = t >> 4;
  const int f = t & 15;
  x0[(size_t)m * ROWW + t] = x[((size_t)b * Mv + m) * FIN + f];
}

// ---------------------------------------------------------------------------
// y = scale * src  (elementwise, float4). scale=0 zeros, scale=-1 gives -x_{k-2}
// ---------------------------------------------------------------------------
__global__ __launch_bounds__(256) void cheb_init(float4* __restrict__ y,
                                                 const float4* __restrict__ src,
                                                 float scale) {
  const size_t i = (size_t)blockIdx.x * 256 + threadIdx.x;   // M*256/4 elems
  float4 s = src[i];
  y[i] = make_float4(scale * s.x, scale * s.y, scale * s.z, scale * s.w);
}

// ---------------------------------------------------------------------------
// Scatter SpMM:  y[rows[e], :] += scale * vals[e] * x[cols[e], :]
// 64 threads (2 waves) per edge, float4 per lane. The edge index is
// wave-uniform: readfirstlane makes rows/cols/vals SMEM scalar loads
// (3 s_loads per wave instead of 96 lane loads). Destination rows are
// L2-resident (50 MB matrix, 192 MB L2); RMW executes at the L2 atomic units.
// ---------------------------------------------------------------------------
__global__ __launch_bounds__(256) void cheb_spmm(float* __restrict__ y,
                                                 const float* __restrict__ x,
                                                 const int* __restrict__ rows,
                                                 const int* __restrict__ cols,
                                                 const float* __restrict__ vals,
                                                 float scale, int nnz) {
  const int t = blockIdx.x * 256 + threadIdx.x;
  const int e = __builtin_amdgcn_readfirstlane(t >> 6);  // wave-uniform edge id
  if (e >= nnz) return;
  const int q = t & 63;               // which float4 of the 256-wide row
  const int r = rows[e];              // scalar (SMEM) loads
  const int c = cols[e];
  const float v = vals[e] * scale;
  const float4 xv = *(const float4*)(x + (size_t)c * ROWW + q * 4);
  float* yp = y + (size_t)r * ROWW + q * 4;
  atomic_add_f32x4(yp, make_float4(v * xv.x, v * xv.y, v * xv.z, v * xv.w));
}

// ---------------------------------------------------------------------------
// Fused Chebyshev GEMM:
//   out[(b*M+m), f1] = sum_{k,f} xs_k[m, b*16+f] * W[f*5+k, f1] + bias[f1]
// K-dim reordered to c' = k*16+f (sum order free). Block = 256 thr = 8 wave32,
// tile = 64 rows x 32 cols, each wave owns one 16x16 D tile, 20 x
// v_wmma_f32_16x16x4_f32 over K=80. LDS strides (84/40) keep the b64 A-frag
// and b32 B-frag loads bank-conflict-free across both lane halves.
// ---------------------------------------------------------------------------
__global__ __launch_bounds__(256) void cheb_gemm(const float* __restrict__ xs,
                                                 const float* __restrict__ W,
                                                 const float* __restrict__ bias,
                                                 float* __restrict__ out) {
  __shared__ float As[64][84];   // [row][c'], 80 cols + pad (stride 84)
  __shared__ float Bs[80][40];   // [c'][n],  32 cols + pad (stride 40)

  const int t  = threadIdx.x;
  const size_t r0 = (size_t)blockIdx.x * 64;         // global row base (b*M+m)

  // --- load + reorder the 80x32 weight tile: Bs[k*16+f][n] = W[f*5+k][n] ---
  for (int i = t; i < 80 * 32; i += 256) {
    const int c = i >> 5, n = i & 31;
    const int k = c >> 4, f = c & 15;
    Bs[c][n] = W[(size_t)(f * Kc + k) * F1 + n];
  }

  // --- load A panel: 64 rows x (5 terms x 16 feats), float4 per (thread,term)
  {
    const int row  = t >> 2;           // 0..63
    const int part = t & 3;            // which float4 of the 16 features
    const size_t r = r0 + row;
    const size_t b = r / Mv;
    const size_t m = r % Mv;
    const float* src = xs + m * ROWW + b * 16 + part * 4;
#pragma unroll
    for (int k = 0; k < Kc; ++k) {
      const float4 v = *(const float4*)(src + (size_t)k * TERM);
      *(float4*)&As[row][k * 16 + part * 4] = v;
    }
  }
  __syncthreads();

  // --- per-wave 16x16 WMMA tile ---
  const int wave = t >> 5;             // 0..7 (wave32)
  const int lane = t & 31;
  const int h    = lane >> 4;          // lane half selects K pair
  const int lm   = lane & 15;
  const int rsub = wave >> 1;          // 0..3 -> 16-row subtile
  const int n0   = (wave & 1) * 16;    // column half
  const int arow = rsub * 16 + lm;

  v8f acc = {0.f, 0.f, 0.f, 0.f, 0.f, 0.f, 0.f, 0.f};
#pragma unroll
  for (int s = 0; s < 20; ++s) {
    const int c0 = s * 4;
    // A fragment: lane half h supplies K = c0+2h, c0+2h+1 (contiguous b64)
    v2f a = *(const v2f*)&As[arow][c0 + 2 * h];
    // B fragment: same K rows, column n0+lm
    v2f bw;
    bw.x = Bs[c0 + 2 * h + 0][n0 + lm];
    bw.y = Bs[c0 + 2 * h + 1][n0 + lm];
    acc = __builtin_amdgcn_wmma_f32_16x16x4_f32(
        /*neg_a=*/false, a, /*neg_b=*/false, bw,
        /*c_mod=*/(short)0, acc, /*reuse_a=*/false, /*reuse_b=*/false);
  }

  const float bvl = bias[n0 + lm];
  const size_t rbase = r0 + (size_t)rsub * 16;
#pragma unroll
  for (int v = 0; v < 8; ++v) {        // D[M=v+8h, N=lm]
    const size_t rr = rbase + v + 8 * h;
    out[rr * F1 + n0 + lm] = acc[v] + bvl;
  }
}

// ---------------------------------------------------------------------------
extern "C" void kernel_launch(void* const* d_in, const int* in_sizes, int n_in,
                              void* d_out, int out_size, void* d_ws, size_t ws_size,
                              hipStream_t stream) {
  const float* x    = (const float*)d_in[0];
  const int*   rows = (const int*)d_in[1];
  const int*   cols = (const int*)d_in[2];
  const float* vals = (const float*)d_in[3];
  const float* W    = (const float*)d_in[4];
  const float* bias = (const float*)d_in[5];
  float*       out  = (float*)d_out;
  float*       xs   = (float*)d_ws;          // 5 terms, each M*256 floats (252 MB)
  const int    nnz  = in_sizes[1];

  const int initBlocks = (int)(TERM / 4 / 256);        // 12288
  const int spmmBlocks = (nnz * 64 + 255) / 256;       // nnz/4
  const int gemmBlocks = (Bv * Mv) / 64;               // 12288

  // T0 = pack(x)
  cheb_pack<<<Mv, 256, 0, stream>>>(x, xs);

  // Chebyshev recurrence: T1 = L@T0 ; Tk = 2*L@T(k-1) - T(k-2)
  for (int k = 1; k < Kc; ++k) {
    float*       y  = xs + (size_t)k * TERM;
    const float* xc = xs + (size_t)(k - 1) * TERM;
    if (k == 1) {
      cheb_init<<<initBlocks, 256, 0, stream>>>((float4*)y, (const float4*)xs, 0.0f);
      cheb_spmm<<<spmmBlocks, 256, 0, stream>>>(y, xc, rows, cols, vals, 1.0f, nnz);
    } else {
      const float* xp = xs + (size_t)(k - 2) * TERM;
      cheb_init<<<initBlocks, 256, 0, stream>>>((float4*)y, (const float4*)xp, -1.0f);
      cheb_spmm<<<spmmBlocks, 256, 0, stream>>>(y, xc, rows, cols, vals, 2.0f, nnz);
    }
  }

  // Fused K=80 GEMM over all 5 stored terms, fp32 WMMA
  cheb_gemm<<<gemmBlocks, 256, 0, stream>>>(xs, W, bias, out);
}